// SelfAttentiveRotation_10531259810082
// MI455X (gfx1250) — compile-verified
//
#include <hip/hip_runtime.h>
#include <math.h>

// MI455X / gfx1250, wave32. Matrix math via V_WMMA_F32_16X16X4_F32 (exact fp32).
// Weight K-panels staged into LDS with the Tensor Data Mover (tensor_load_to_lds,
// TENSORcnt), double-buffered. Workspace requirement: 192 MB (q/k/v intermediates).

typedef float        v2f __attribute__((ext_vector_type(2)));
typedef float        v8f __attribute__((ext_vector_type(8)));
typedef unsigned int u32;
typedef u32          v4u __attribute__((ext_vector_type(4)));
typedef u32          v8u __attribute__((ext_vector_type(8)));

#define BATCH 1024
#define FNUM  64
#define DIN   256
#define DOUT  256
#define NHEAD 8
#define DH    32
#define KPANEL 64
#define LN_EPS 1e-5f

// ---- WMMA fragment helpers (16x16x4 fp32) -------------------------------
__device__ __forceinline__ v2f load_a_frag(const float* base, int lda, int k0, int lane) {
  int row = lane & 15;
  int kb  = k0 + ((lane >> 4) << 1);
  const float* p = base + row * lda + kb;
  v2f r; r.x = p[0]; r.y = p[1];
  return r;
}
__device__ __forceinline__ v2f load_b_frag_rm(const float* base, int ldb, int k0, int n0, int lane) {
  int col = n0 + (lane & 15);
  int kb  = k0 + ((lane >> 4) << 1);
  v2f r; r.x = base[kb * ldb + col]; r.y = base[(kb + 1) * ldb + col];
  return r;
}
__device__ __forceinline__ v2f load_b_frag_tr(const float* base, int ldw, int k0, int n0, int lane) {
  int col = n0 + (lane & 15);
  int kb  = k0 + ((lane >> 4) << 1);
  const float* p = base + col * ldw + kb;
  v2f r; r.x = p[0]; r.y = p[1];
  return r;
}
__device__ __forceinline__ v8f wmma4(v2f a, v2f b, v8f c) {
  return __builtin_amdgcn_wmma_f32_16x16x4_f32(false, a, false, b, (short)0, c, false, false);
}
__device__ __forceinline__ v8f vzero8() {
  v8f z = {0.f, 0.f, 0.f, 0.f, 0.f, 0.f, 0.f, 0.f};
  return z;
}

// ---- Tensor Data Mover: 2D tile (rows x cols fp32) global -> LDS ----------
// D# per CDNA5 ISA ch.8: group0 = {count|flags, lds_addr, gaddr_lo, gaddr_hi|type=2},
// group1 = {mask/data_size, dims/tiles/stride}, groups 2/3 zero (2-D tile).
__device__ __forceinline__ void tdm_load_2d(const float* gptr, const float* lptr,
                                            int rows, int cols, int stride_elems) {
  unsigned long long ga = (unsigned long long)(uintptr_t)gptr;
  u32 la = (u32)(uintptr_t)lptr;          // low 32 bits of flat addr = LDS byte offset
  v4u g0;
  g0[0] = 1u;                              // count=1, user descriptor
  g0[1] = la;                              // lds_addr
  g0[2] = (u32)ga;                         // global_addr[31:0]
  g0[3] = (u32)(ga >> 32) | (2u << 30);    // global_addr[56:32] | type=2 ("image")
  v8u g1;
  g1[0] = 2u << 16;                                            // data_size=4B, mask=0
  g1[1] = ((u32)cols & 0xffffu) << 16;                         // tensor_dim0[15:0]
  g1[2] = (((u32)cols >> 16) & 0xffffu) |
          (((u32)rows & 0xffffu) << 16);                       // dim0[31:16] | dim1[15:0]
  g1[3] = (((u32)rows >> 16) & 0xffffu) |
          (((u32)cols & 0xffffu) << 16);                       // dim1[31:16] | tile_dim0
  g1[4] = (u32)rows & 0xffffu;                                 // tile_dim1 (tile_dim2=0)
  g1[5] = (u32)stride_elems;                                   // tensor_dim0_stride lo32
  g1[6] = 0u;
  g1[7] = 0u;
  v4u gz; gz[0] = 0u; gz[1] = 0u; gz[2] = 0u; gz[3] = 0u;
  asm volatile("tensor_load_to_lds %0, %1, %2, %3"
               :: "s"(g0), "s"(g1), "s"(gz), "s"(gz)
               : "memory");
}

// ---- Kernel 1: q/k/v projections --------------------------------------------
// grid (16, 64, 3): x = 64-row batch tile, y = f, z = {Q,K,V}. 256 thr = 8 waves.
// Weight K-panels (64x256 fp32 = 64KB) TDM-staged into LDS, double buffered.
__global__ __launch_bounds__(256)
void sar_qkv_proj(const float* __restrict__ theta, const float* __restrict__ Qw,
                  const float* __restrict__ Kw, const float* __restrict__ Vw,
                  float* __restrict__ q_ws, float* __restrict__ k_ws, float* __restrict__ v_ws)
{
  __shared__ __align__(1024) float wbuf[2][KPANEL * DOUT];   // 2 x 64 KB

  const int mblk = blockIdx.x;
  const int f    = blockIdx.y;
  const int z    = blockIdx.z;
  const float* W = (z == 0 ? Qw : (z == 1 ? Kw : Vw)) + (size_t)f * DIN * DOUT;
  float* outp    = (z == 0 ? q_ws : (z == 1 ? k_ws : v_ws));

  const int tid  = threadIdx.x, lane = tid & 31, wid = tid >> 5;
  const int mtile  = wid >> 1;            // 0..3
  const int n0base = (wid & 1) * 128;     // 8 N-tiles of 16

  const int m0 = mblk * 64 + mtile * 16;  // batch-row base
  const float* abase = theta + (size_t)m0 * (FNUM * DIN) + (size_t)f * DIN;

  if (wid == 0) tdm_load_2d(W, &wbuf[0][0], KPANEL, DOUT, DOUT);

  v8f acc[8];
  #pragma unroll
  for (int t = 0; t < 8; ++t) acc[t] = vzero8();

  #pragma unroll
  for (int p = 0; p < DIN / KPANEL; ++p) {
    if (wid == 0) {
      if (p < DIN / KPANEL - 1) {
        tdm_load_2d(W + (size_t)(p + 1) * KPANEL * DOUT, &wbuf[(p + 1) & 1][0],
                    KPANEL, DOUT, DOUT);
        __builtin_amdgcn_s_wait_tensorcnt(1);
      } else {
        __builtin_amdgcn_s_wait_tensorcnt(0);
      }
    }
    __syncthreads();
    const float* bp = &wbuf[p & 1][0];
    for (int kk = 0; kk < KPANEL; kk += 4) {
      v2f a = load_a_frag(abase, FNUM * DIN, p * KPANEL + kk, lane);
      #pragma unroll
      for (int t = 0; t < 8; ++t) {
        v2f b = load_b_frag_rm(bp, DOUT, kk, n0base + t * 16, lane);
        acc[t] = wmma4(a, b, acc[t]);
      }
    }
    __syncthreads();
  }

  const int hi = lane >> 4, col = lane & 15;
  #pragma unroll
  for (int t = 0; t < 8; ++t) {
    int ncol = n0base + t * 16 + col;
    #pragma unroll
    for (int r = 0; r < 8; ++r) {
      int brow = m0 + r + 8 * hi;
      outp[(size_t)brow * (FNUM * DOUT) + (size_t)f * DOUT + ncol] = acc[t][r];
    }
  }
}

// ---- Kernel 2: residual linear + rotation attention + LayerNorm -------------
// One block per batch element b, 256 threads = 8 waves. W_lin (256KB) stays in L2.
__global__ __launch_bounds__(256)
void sar_attn_ln(const float* __restrict__ theta, const float* __restrict__ W_lin,
                 const float* __restrict__ rot_w, const float* __restrict__ gamma,
                 const float* __restrict__ beta,
                 const float* __restrict__ q_ws, const float* __restrict__ k_ws,
                 const float* __restrict__ v_ws, float* __restrict__ out)
{
  __shared__ float res[64 * 260];   // residual accumulator, padded stride
  __shared__ float cq[64 * 36];
  __shared__ float sq[64 * 36];
  __shared__ float ck[64 * 36];
  __shared__ float sk[64 * 36];
  __shared__ float vv[64 * 36];
  __shared__ float Ssh[64 * 68];
  __shared__ float rot_s[DH];

  const int b   = blockIdx.x;
  const int tid = threadIdx.x, lane = tid & 31, wid = tid >> 5;
  if (tid < DH) rot_s[tid] = rot_w[tid];

  const float* abase0 = theta + (size_t)b * (FNUM * DIN);

  // --- linear residual: res = theta[b] @ W_lin^T  (contract over d) ---
  {
    const int mtile  = wid >> 1;
    const int n0base = (wid & 1) * 128;
    const float* abase = abase0 + (size_t)mtile * 16 * DIN;
    v8f acc[8];
    #pragma unroll
    for (int t = 0; t < 8; ++t) acc[t] = vzero8();
    for (int k0 = 0; k0 < DIN; k0 += 4) {
      v2f a = load_a_frag(abase, DIN, k0, lane);
      #pragma unroll
      for (int t = 0; t < 8; ++t) {
        v2f bf = load_b_frag_tr(W_lin, DIN, k0, n0base + t * 16, lane);
        acc[t] = wmma4(a, bf, acc[t]);
      }
    }
    const int hi = lane >> 4, col = lane & 15;
    #pragma unroll
    for (int t = 0; t < 8; ++t)
      #pragma unroll
      for (int r = 0; r < 8; ++r)
        res[(mtile * 16 + r + 8 * hi) * 260 + n0base + t * 16 + col] = acc[t][r];
  }
  __syncthreads();

  const size_t bofs = (size_t)b * (FNUM * DOUT);
  for (int h = 0; h < NHEAD; ++h) {
    // load per-head slices, apply cos/sin (+rot_w on q side) at load time
    for (int i = tid; i < 64 * DH; i += 256) {
      int frow = i >> 5;
      int c    = i & 31;
      size_t gofs = bofs + (size_t)frow * DOUT + h * DH + c;
      float rw = rot_s[c];
      float qv = q_ws[gofs];
      float kv = k_ws[gofs];
      cq[frow * 36 + c] = __cosf(qv) * rw;
      sq[frow * 36 + c] = __sinf(qv) * rw;
      ck[frow * 36 + c] = __cosf(kv);
      sk[frow * 36 + c] = __sinf(kv);
      vv[frow * 36 + c] = v_ws[gofs];
    }
    __syncthreads();

    // logits = cq@ck^T + sq@sk^T ; sigmoid -> Ssh   (4x4 tiles, 2 per wave)
    {
      const int mtile = wid >> 1;
      const int nt0   = (wid & 1) * 2;
      v8f accL[2];
      accL[0] = vzero8(); accL[1] = vzero8();
      for (int k0 = 0; k0 < DH; k0 += 4) {
        v2f ac = load_a_frag(cq + mtile * 16 * 36, 36, k0, lane);
        v2f as = load_a_frag(sq + mtile * 16 * 36, 36, k0, lane);
        #pragma unroll
        for (int t = 0; t < 2; ++t) {
          v2f bc = load_b_frag_tr(ck, 36, k0, (nt0 + t) * 16, lane);
          v2f bs = load_b_frag_tr(sk, 36, k0, (nt0 + t) * 16, lane);
          accL[t] = wmma4(ac, bc, accL[t]);
          accL[t] = wmma4(as, bs, accL[t]);
        }
      }
      const int hi = lane >> 4, col = lane & 15;
      #pragma unroll
      for (int t = 0; t < 2; ++t)
        #pragma unroll
        for (int r = 0; r < 8; ++r) {
          float x = accL[t][r];
          Ssh[(mtile * 16 + r + 8 * hi) * 68 + (nt0 + t) * 16 + col] =
              1.f / (1.f + __expf(-x));
        }
    }
    __syncthreads();

    // out_h = Ssh @ vv, accumulate into res[:, h*32 : h*32+32]  (4x2 tiles, 1/wave)
    {
      const int mtile = wid >> 1;
      const int ntile = wid & 1;
      v8f acc = vzero8();
      for (int k0 = 0; k0 < 64; k0 += 4) {
        v2f a  = load_a_frag(Ssh + mtile * 16 * 68, 68, k0, lane);
        v2f bf = load_b_frag_rm(vv, 36, k0, ntile * 16, lane);
        acc = wmma4(a, bf, acc);
      }
      const int hi = lane >> 4, col = lane & 15;
      #pragma unroll
      for (int r = 0; r < 8; ++r) {
        int rrow = mtile * 16 + r + 8 * hi;
        res[rrow * 260 + h * DH + ntile * 16 + col] += acc[r];
      }
    }
    __syncthreads();
  }

  // --- LayerNorm over 256, one wave per row ---
  for (int frow = wid; frow < 64; frow += 8) {
    float s = 0.f, s2 = 0.f;
    float vals[8];
    #pragma unroll
    for (int j = 0; j < 8; ++j) {
      float x = res[frow * 260 + lane + 32 * j];
      vals[j] = x; s += x; s2 += x * x;
    }
    #pragma unroll
    for (int off = 16; off > 0; off >>= 1) {
      s  += __shfl_xor(s,  off, 32);
      s2 += __shfl_xor(s2, off, 32);
    }
    float mu  = s * (1.f / 256.f);
    float var = s2 * (1.f / 256.f) - mu * mu;
    float rs  = rsqrtf(var + LN_EPS);
    #pragma unroll
    for (int j = 0; j < 8; ++j) {
      int c = lane + 32 * j;
      out[bofs + (size_t)frow * DOUT + c] = (vals[j] - mu) * rs * gamma[c] + beta[c];
    }
  }
}

extern "C" void kernel_launch(void* const* d_in, const int* in_sizes, int n_in,
                              void* d_out, int out_size, void* d_ws, size_t ws_size,
                              hipStream_t stream) {
  (void)in_sizes; (void)n_in; (void)out_size; (void)ws_size;
  const float* theta = (const float*)d_in[0];
  const float* Qw    = (const float*)d_in[1];
  const float* Kw    = (const float*)d_in[2];
  const float* Vw    = (const float*)d_in[3];
  const float* W_lin = (const float*)d_in[4];
  const float* rot_w = (const float*)d_in[5];
  const float* gamma = (const float*)d_in[6];
  const float* beta  = (const float*)d_in[7];
  float* out = (float*)d_out;

  const size_t per = (size_t)BATCH * FNUM * DOUT;  // 16.7M floats = 64 MB
  float* q_ws = (float*)d_ws;
  float* k_ws = q_ws + per;
  float* v_ws = k_ws + per;

  dim3 g1(BATCH / 64, FNUM, 3);
  sar_qkv_proj<<<g1, 256, 0, stream>>>(theta, Qw, Kw, Vw, q_ws, k_ws, v_ws);
  sar_attn_ln<<<BATCH, 256, 0, stream>>>(theta, W_lin, rot_w, gamma, beta,
                                         q_ws, k_ws, v_ws, out);
}